// GlobalCluster_1434519077361
// MI455X (gfx1250) — compile-verified
//
#include <hip/hip_runtime.h>
#include <math.h>

// fp32 WMMA fragments: A/B = 2 VGPRs (16x4 / 4x16 f32), C/D = 8 VGPRs (16x16 f32)
typedef __attribute__((ext_vector_type(2))) float v2f;
typedef __attribute__((ext_vector_type(8))) float v8f;
typedef __attribute__((ext_vector_type(4))) unsigned int u32x4;
typedef __attribute__((ext_vector_type(8))) int i32x8;
typedef __attribute__((ext_vector_type(4))) int i32x4;

#define EPSN 1e-12f
// Staged LDS rows: 32 data dwords + 2 pad dwords (inserted by TDM pad logic).
// stride 34 dwords -> rows stay 8B-aligned (ds_load_b64) and 34 mod 64 banks
// spreads the 16 row-lanes across distinct banks (conflict-free).
#define ROWSTR 34

__device__ __forceinline__ v8f wmma4(v2f a, v2f b, v8f c) {
    // D = A(16x4) * B(4x16) + C, fp32
    return __builtin_amdgcn_wmma_f32_16x16x4_f32(false, a, false, b, (short)0, c, false, false);
}

// ---------------------------------------------------------------------------
// Tensor Data Mover: 2D tile (4-byte elements) global -> LDS.
// D# packing per CDNA5 ISA ch.8: group0 {count=1, lds_addr, global_addr, type=2},
// group1 {data_size=4B, pad_enable, pad_interval=32dw, pad_amount=2dw,
//         tensor dims == tile dims (no OOB), stride0}.
// 6-arg builtin form (clang-23 toolchain); groups 2/3 + trailing group zero
// for a 2D tile. Tracked by TENSORcnt; caller waits with s_wait_tensorcnt.
// ---------------------------------------------------------------------------
__device__ __forceinline__ void tdm_load_2d(const void* gsrc, unsigned ldsByteOff,
                                            unsigned tileD0, unsigned tileD1,
                                            unsigned strideD0) {
    unsigned long long ga = (unsigned long long)(uintptr_t)gsrc;
    u32x4 g0;
    g0[0] = 1u;                                                   // count=1 (valid user D#)
    g0[1] = ldsByteOff;                                           // lds_addr [63:32]
    g0[2] = (unsigned)ga;                                         // global_addr[31:0]
    g0[3] = (unsigned)((ga >> 32) & 0x1ffffffull) | 0x80000000u;  // addr[56:32] | type=2
    i32x8 g1;
    g1[0] = (int)((2u << 16)      // data_size = 4 bytes
                | (1u << 20)      // pad_enable
                | (4u << 22)      // pad_interval code 4 = every 32 dwords
                | (1u << 25));    // pad_amount  code 1 = 2 dwords
    g1[1] = (int)((tileD0 & 0xffffu) << 16);                                  // tensor_dim0 lo
    g1[2] = (int)(((tileD0 >> 16) & 0xffffu) | ((tileD1 & 0xffffu) << 16));   // dim0 hi | dim1 lo
    g1[3] = (int)(((tileD1 >> 16) & 0xffffu) | ((tileD0 & 0xffffu) << 16));   // dim1 hi | tile_dim0
    g1[4] = (int)(tileD1 & 0xffffu);                                          // tile_dim1 | tile_dim2=0
    g1[5] = (int)strideD0;                                                    // tensor_dim0_stride
    g1[6] = 0;
    g1[7] = 0;
    i32x4 z4 = {0, 0, 0, 0};
    i32x8 z8 = {0, 0, 0, 0, 0, 0, 0, 0};
    __builtin_amdgcn_tensor_load_to_lds(g0, g1, z4, z4, z8, 0);
}

// ---------------------------------------------------------------------------
// Kernel A: x0p = x0 @ W0^T + b0, per-row L2-normalize each 32-channel group,
// store as xn[mb][lrow][0..31]. WG = 64 rows x 64 cols (4 waves x 4 acc).
// A and B k-slabs staged in LDS by the TDM.
// ---------------------------------------------------------------------------
__global__ void k_proj_x0(const float* __restrict__ x0, const float* __restrict__ W0,
                          const float* __restrict__ b0, float* __restrict__ xn,
                          int C, int H, int l) {
    extern __shared__ float smem[];
    const unsigned ldsBase = (unsigned)__builtin_amdgcn_groupstaticsize();
    float* As = smem;                   // 64 x ROWSTR
    float* Bs = smem + 64 * ROWSTR;     // 64 x ROWSTR
    const unsigned AsOff = ldsBase;
    const unsigned BsOff = ldsBase + 64u * ROWSTR * 4u;

    const int ty = threadIdx.y;
    const int lane = threadIdx.x;
    const int l15 = lane & 15;
    const int hi  = lane >> 4;
    const int colBlocks = H >> 6;             // 4
    const int cb = blockIdx.x % colBlocks;    // 64-col block
    const int rb = blockIdx.x / colBlocks;    // 64-row block

    v8f acc[4] = {};
    for (int ks = 0; ks < C; ks += 32) {
        __syncthreads();                       // previous slab fully consumed
        if (ty == 0) {
            tdm_load_2d(x0 + (size_t)(rb * 64) * C + ks, AsOff, 32, 64, (unsigned)C);
            tdm_load_2d(W0 + (size_t)(cb * 64) * C + ks, BsOff, 32, 64, (unsigned)C);
            __builtin_amdgcn_s_wait_tensorcnt(0);
        }
        __syncthreads();                       // slab visible to all waves
        const float* Ar = As + (ty * 16 + l15) * ROWSTR + 2 * hi;
        const float* Br = Bs + l15 * ROWSTR + 2 * hi;
#pragma unroll
        for (int kk = 0; kk < 32; kk += 4) {
            v2f a = *(const v2f*)(Ar + kk);
#pragma unroll
            for (int c = 0; c < 4; ++c)
                acc[c] = wmma4(a, *(const v2f*)(Br + c * 16 * ROWSTR + kk), acc[c]);
        }
    }

    float bias[4];
#pragma unroll
    for (int c = 0; c < 4; ++c) bias[c] = b0[cb * 64 + c * 16 + l15];

#pragma unroll
    for (int r = 0; r < 8; ++r) {
        const int gr   = rb * 64 + ty * 16 + r + 8 * hi;   // global row in [0, n*l)
        const int nn   = gr / l;
        const int lrow = gr - nn * l;
#pragma unroll
        for (int g = 0; g < 2; ++g) {          // two 32-col normalize groups per wave
            float e0 = acc[2 * g][r] + bias[2 * g];
            float e1 = acc[2 * g + 1][r] + bias[2 * g + 1];
            float sq = e0 * e0 + e1 * e1;
            sq += __shfl_xor(sq, 1);
            sq += __shfl_xor(sq, 2);
            sq += __shfl_xor(sq, 4);
            sq += __shfl_xor(sq, 8);
            const float inv = 1.0f / fmaxf(sqrtf(sq), EPSN);
            const int f = cb * 2 + g;
            const size_t base = (((size_t)(nn * 8 + f) * l) + lrow) * 32;
            xn[base + l15]      = e0 * inv;
            xn[base + 16 + l15] = e1 * inv;
        }
    }
}

// ---------------------------------------------------------------------------
// Kernel B: c1 = center1 @ W1^T + b1 -> normalized c_point (cn) + raw c_value
// (cv). Small GEMM (0.27 GFLOP): plain global-load WMMA, one wave = 16x64.
// ---------------------------------------------------------------------------
__global__ void k_proj_c(const float* __restrict__ center1, const float* __restrict__ W1,
                         const float* __restrict__ b1, float* __restrict__ cn,
                         float* __restrict__ cv, int C, int H, int s) {
    const int wave = blockIdx.x * blockDim.y + threadIdx.y;
    const int lane = threadIdx.x;
    const int l15 = lane & 15;
    const int hi  = lane >> 4;
    const int groups = H >> 5;                 // 8 (= 2H/64)
    const int rowTile = wave / groups;
    const int f       = wave % groups;

    const float* Abase = center1 + (size_t)(rowTile * 16 + l15) * C;
    const float* Bb0 = W1 + (size_t)(f * 64 +      l15) * C;
    const float* Bb1 = W1 + (size_t)(f * 64 + 16 + l15) * C;
    const float* Bb2 = W1 + (size_t)(f * 64 + 32 + l15) * C;
    const float* Bb3 = W1 + (size_t)(f * 64 + 48 + l15) * C;

    v8f d0 = {}; v8f d1 = {}; v8f d2 = {}; v8f d3 = {};
    for (int k = 0; k < C; k += 4) {
        v2f a = *(const v2f*)(Abase + k + 2 * hi);
        d0 = wmma4(a, *(const v2f*)(Bb0 + k + 2 * hi), d0);
        d1 = wmma4(a, *(const v2f*)(Bb1 + k + 2 * hi), d1);
        d2 = wmma4(a, *(const v2f*)(Bb2 + k + 2 * hi), d2);
        d3 = wmma4(a, *(const v2f*)(Bb3 + k + 2 * hi), d3);
    }

    const float bi0 = b1[f * 64 + l15];
    const float bi1 = b1[f * 64 + 16 + l15];
    const float bi2 = b1[f * 64 + 32 + l15];
    const float bi3 = b1[f * 64 + 48 + l15];

#pragma unroll
    for (int r = 0; r < 8; ++r) {
        float e0 = d0[r] + bi0;
        float e1 = d1[r] + bi1;
        float e2 = d2[r] + bi2;
        float e3 = d3[r] + bi3;
        float sq = e0 * e0 + e1 * e1;
        sq += __shfl_xor(sq, 1);
        sq += __shfl_xor(sq, 2);
        sq += __shfl_xor(sq, 4);
        sq += __shfl_xor(sq, 8);
        const float inv = 1.0f / fmaxf(sqrtf(sq), EPSN);
        const int gr   = rowTile * 16 + r + 8 * hi;   // in [0, n*s)
        const int nn   = gr / s;
        const int srow = gr - nn * s;
        const size_t base = (((size_t)(nn * 8 + f) * s) + srow) * 32;
        cn[base + l15]      = e0 * inv;
        cn[base + 16 + l15] = e1 * inv;
        cv[base + l15]      = e2;
        cv[base + 16 + l15] = e3;
    }
}

// ---------------------------------------------------------------------------
// Kernel C: sim = xn . cn (K=32) -> sigmoid -> max/argmax over s -> gather
// c_value -> scale -> scatter into disp[n][lrow][f*32+c]. sim never hits
// memory. WG stages cn[mb] (256x32) + a 64-row xn slab via two TDM loads.
// ---------------------------------------------------------------------------
__global__ void k_sim_dispatch(const float* __restrict__ xn, const float* __restrict__ cn,
                               const float* __restrict__ cv,
                               const float* __restrict__ alpha, const float* __restrict__ beta,
                               float* __restrict__ disp, int l, int s, int H) {
    extern __shared__ float smem[];
    const unsigned ldsBase = (unsigned)__builtin_amdgcn_groupstaticsize();
    float* Cs = smem;                        // s  x ROWSTR   (cn[mb])
    float* As = smem + (size_t)s * ROWSTR;   // 64 x ROWSTR   (xn rows)
    const unsigned CsOff = ldsBase;
    const unsigned AsOff = ldsBase + (unsigned)(s * ROWSTR * 4);

    const int ty = threadIdx.y;
    const int lane = threadIdx.x;
    const int l15 = lane & 15;
    const int hi  = lane >> 4;
    const int ltBlocks = l >> 6;               // 64
    const int mb  = blockIdx.x / ltBlocks;
    const int ltb = blockIdx.x % ltBlocks;     // 64-row l block

    if (ty == 0) {
        tdm_load_2d(cn + (size_t)mb * s * 32,               CsOff, 32, (unsigned)s, 32);
        tdm_load_2d(xn + ((size_t)mb * l + ltb * 64) * 32,  AsOff, 32, 64,          32);
        __builtin_amdgcn_s_wait_tensorcnt(0);
    }
    __syncthreads();

    // full A fragment set for this wave's 16 rows (K=32 -> 8 k-steps)
    v2f afr[8];
    const float* Ar = As + (ty * 16 + l15) * ROWSTR + 2 * hi;
#pragma unroll
    for (int kk = 0; kk < 8; ++kk) afr[kk] = *(const v2f*)(Ar + kk * 4);

    const float al = alpha[0];
    const float be = beta[0];

    float maxv[8]; int maxi[8];
#pragma unroll
    for (int r = 0; r < 8; ++r) { maxv[r] = -INFINITY; maxi[r] = 0; }

    for (int st = 0; st < (s >> 4); ++st) {
        const float* Br = Cs + (st * 16 + l15) * ROWSTR + 2 * hi;
        v8f acc = {};
#pragma unroll
        for (int kk = 0; kk < 8; ++kk)
            acc = wmma4(afr[kk], *(const v2f*)(Br + kk * 4), acc);
#pragma unroll
        for (int r = 0; r < 8; ++r) {
            const float t  = al * acc[r] + be;
            const float sg = 1.0f / (1.0f + __expf(-t));
            if (sg > maxv[r]) { maxv[r] = sg; maxi[r] = st * 16 + l15; }
        }
    }

    const int nn = mb >> 3;
    const int f  = mb & 7;
#pragma unroll
    for (int r = 0; r < 8; ++r) {
        float mv = maxv[r]; int mi = maxi[r];
#pragma unroll
        for (int x = 1; x < 16; x <<= 1) {
            const float ov = __shfl_xor(mv, x);
            const int   oi = __shfl_xor(mi, x);
            if (ov > mv || (ov == mv && oi < mi)) { mv = ov; mi = oi; }  // first-max tiebreak
        }
        const int lrow = ltb * 64 + ty * 16 + r + 8 * hi;
        const float* cvrow = cv + ((size_t)mb * s + mi) * 32;
        v2f val;
        val.x = mv * cvrow[2 * l15];
        val.y = mv * cvrow[2 * l15 + 1];
        const size_t obase = (((size_t)nn * l + lrow) * H) + f * 32;
        *(v2f*)(disp + obase + 2 * l15) = val;
    }
}

// ---------------------------------------------------------------------------
// Kernel D: out = disp @ Wm^T + bm. Same 64x64 TDM-staged blocking as kernel A.
// ---------------------------------------------------------------------------
__global__ void k_out(const float* __restrict__ disp, const float* __restrict__ Wm,
                      const float* __restrict__ bm, float* __restrict__ out,
                      int H, int C) {
    extern __shared__ float smem[];
    const unsigned ldsBase = (unsigned)__builtin_amdgcn_groupstaticsize();
    float* As = smem;                   // 64 x ROWSTR
    float* Bs = smem + 64 * ROWSTR;     // 64 x ROWSTR
    const unsigned AsOff = ldsBase;
    const unsigned BsOff = ldsBase + 64u * ROWSTR * 4u;

    const int ty = threadIdx.y;
    const int lane = threadIdx.x;
    const int l15 = lane & 15;
    const int hi  = lane >> 4;
    const int colBlocks = C >> 6;             // 4
    const int cb = blockIdx.x % colBlocks;
    const int rb = blockIdx.x / colBlocks;

    v8f acc[4] = {};
    for (int ks = 0; ks < H; ks += 32) {
        __syncthreads();
        if (ty == 0) {
            tdm_load_2d(disp + (size_t)(rb * 64) * H + ks, AsOff, 32, 64, (unsigned)H);
            tdm_load_2d(Wm   + (size_t)(cb * 64) * H + ks, BsOff, 32, 64, (unsigned)H);
            __builtin_amdgcn_s_wait_tensorcnt(0);
        }
        __syncthreads();
        const float* Ar = As + (ty * 16 + l15) * ROWSTR + 2 * hi;
        const float* Br = Bs + l15 * ROWSTR + 2 * hi;
#pragma unroll
        for (int kk = 0; kk < 32; kk += 4) {
            v2f a = *(const v2f*)(Ar + kk);
#pragma unroll
            for (int c = 0; c < 4; ++c)
                acc[c] = wmma4(a, *(const v2f*)(Br + c * 16 * ROWSTR + kk), acc[c]);
        }
    }

#pragma unroll
    for (int r = 0; r < 8; ++r) {
        const int gr = rb * 64 + ty * 16 + r + 8 * hi;
#pragma unroll
        for (int c = 0; c < 4; ++c)
            out[(size_t)gr * C + cb * 64 + c * 16 + l15] = acc[c][r] + bm[cb * 64 + c * 16 + l15];
    }
}

// ---------------------------------------------------------------------------
extern "C" void kernel_launch(void* const* d_in, const int* in_sizes, int n_in,
                              void* d_out, int out_size, void* d_ws, size_t ws_size,
                              hipStream_t stream) {
    const float* x0      = (const float*)d_in[0];
    const float* center1 = (const float*)d_in[1];
    const float* W0      = (const float*)d_in[2];
    const float* b0      = (const float*)d_in[3];
    const float* W1      = (const float*)d_in[4];
    const float* b1      = (const float*)d_in[5];
    const float* Wm      = (const float*)d_in[6];
    const float* bm      = (const float*)d_in[7];
    const float* alpha   = (const float*)d_in[8];
    const float* beta    = (const float*)d_in[9];
    float* out = (float*)d_out;

    const int n = 4;
    const int H = in_sizes[3];               // 256
    const int C = in_sizes[2] / H;           // 256
    const int nl = in_sizes[0] / C;          // 16384
    const int l  = nl / n;                   // 4096
    const int ns = in_sizes[1] / C;          // 1024
    const int s  = ns / n;                   // 256
    const int m  = n * 8;                    // 32

    float* ws = (float*)d_ws;
    float* xn   = ws;                                    // m*l*32
    float* cn   = xn + (size_t)m * l * 32;               // m*s*32
    float* cv   = cn + (size_t)m * s * 32;               // m*s*32
    float* disp = cv + (size_t)m * s * 32;               // n*l*H

    const dim3 blk(32, 4);
    const size_t shAD = (size_t)(64 + 64) * ROWSTR * 4;     // 17408 B
    const size_t shC  = ((size_t)s + 64) * ROWSTR * 4;      // 43520 B

    const int blocksA = (nl / 64) * (H / 64);               // 1024
    const int blocksB = ((ns / 16) * (H / 32)) / 4;         // 128
    const int blocksC = m * (l / 64);                       // 2048
    const int blocksD = (nl / 64) * (C / 64);               // 1024

    k_proj_x0<<<blocksA, blk, shAD, stream>>>(x0, W0, b0, xn, C, H, l);
    k_proj_c<<<blocksB, blk, 0, stream>>>(center1, W1, b1, cn, cv, C, H, s);
    k_sim_dispatch<<<blocksC, blk, shC, stream>>>(xn, cn, cv, alpha, beta, disp, l, s, H);
    k_out<<<blocksD, blk, shAD, stream>>>(disp, Wm, bm, out, H, C);
}